// BasicRGCN_19172734010142
// MI455X (gfx1250) — compile-verified
//
#include <hip/hip_runtime.h>

#define N_ENT 4096
#define N_REL 4
#define FDIM  128
#define BATCH 8192
#define CHUNK 1024

typedef __attribute__((ext_vector_type(2))) float v2f;
typedef __attribute__((ext_vector_type(8))) float v8f;

// ---------------------------------------------------------------------------
// K1: sparse aggregation + degree normalization.
// One block per (r, n) row of adj. Phase 1: 128 threads stream-scan a 1024-col
// chunk of the adj row with float4 (global_load_b128: 512B/wave/instr against
// the 23.3 TB/s HBM stream), pushing nonzeros into an LDS list. Phase 2:
// threads (mapped to feature dim f) accumulate val*feat[m][f] for each listed
// nonzero (feat is L2-resident: 2MB << 192MB L2).
// Writes tmp[r][n][f] = agg * inv_deg  (inv_deg = 1/count_nonzero, 0 -> 1).
// ---------------------------------------------------------------------------
__global__ __launch_bounds__(128) void k_agg(const float* __restrict__ adj,
                                             const float* __restrict__ feat,
                                             float* __restrict__ tmp) {
    const int row = blockIdx.x;              // r*N_ENT + n
    const float* __restrict__ arow = adj + ((size_t)row << 12);
    const int f = threadIdx.x;               // 0..127 feature lane

    __shared__ int   s_idx[CHUNK];
    __shared__ float s_val[CHUNK];
    __shared__ int   s_cnt;

    float acc = 0.0f;
    int deg = 0;

    for (int c0 = 0; c0 < N_ENT; c0 += CHUNK) {
        if (threadIdx.x == 0) s_cnt = 0;
        __syncthreads();
        // stream-scan this chunk as float4; prefetch next chunk
        const float4* __restrict__ arow4 = (const float4*)(arow + c0);
        #pragma unroll
        for (int it = 0; it < CHUNK / (128 * 4); ++it) {      // 2 iterations
            const int q = it * 128 + threadIdx.x;             // float4 slot
            if (c0 + CHUNK < N_ENT)
                __builtin_prefetch((const float4*)(arow + c0 + CHUNK) + q, 0, 0);
            float4 a = arow4[q];
            const int col = c0 + q * 4;
            if (a.x != 0.0f) { int p = atomicAdd(&s_cnt, 1); s_idx[p] = col;     s_val[p] = a.x; }
            if (a.y != 0.0f) { int p = atomicAdd(&s_cnt, 1); s_idx[p] = col + 1; s_val[p] = a.y; }
            if (a.z != 0.0f) { int p = atomicAdd(&s_cnt, 1); s_idx[p] = col + 2; s_val[p] = a.z; }
            if (a.w != 0.0f) { int p = atomicAdd(&s_cnt, 1); s_idx[p] = col + 3; s_val[p] = a.w; }
        }
        __syncthreads();
        const int cnt = s_cnt;
        deg += cnt;
        for (int j = 0; j < cnt; ++j) {
            acc = fmaf(s_val[j], feat[((size_t)s_idx[j] << 7) + f], acc);
        }
        __syncthreads();
    }
    const float inv = (deg == 0) ? 1.0f : (1.0f / (float)deg);
    tmp[((size_t)row << 7) + f] = inv * acc;
}

// ---------------------------------------------------------------------------
// K2: h[n][o] = sigmoid( sum_r  tmp[r][n][:] . W[r][o][:] )
// fp32 WMMA GEMM: 16x16 C tile per wave, K accumulated over 4 relations x 128.
// A fragment (16x4 f32): VGPR0 = {K0 | K2}, VGPR1 = {K1 | K3} per ISA 7.12.2.
// B fragment (4x16): B[k][ncol] = W[r][o0+ncol][k]  (transpose-on-load).
// C tile: VGPR i -> row (i + 8*half), col l16.
// Block = 256 threads = 8 waves; wave w owns o-tile w. grid = N_ENT/16.
// ---------------------------------------------------------------------------
__global__ __launch_bounds__(256) void k_gemm(const float* __restrict__ tmp, // [R][N][F]
                                              const float* __restrict__ W,   // [R][F][F]
                                              float* __restrict__ hOut) {    // [N][F]
    const int wave = threadIdx.x >> 5;       // wave32
    const int lane = threadIdx.x & 31;
    const int half = lane >> 4;              // 0: lanes 0-15, 1: lanes 16-31
    const int l16  = lane & 15;
    const int n0   = blockIdx.x * 16;
    const int o0   = wave * 16;

    v8f c = {};
    for (int r = 0; r < N_REL; ++r) {
        const float* __restrict__ A =
            tmp + (((size_t)r * N_ENT + (n0 + l16)) << 7);   // row n0+l16 of agg_r
        const float* __restrict__ B =
            W + (((size_t)r * FDIM + (o0 + l16)) << 7);      // row o0+l16 of W_r
        for (int k = 0; k < FDIM; k += 4) {
            v2f a, b;
            a.x = A[k + 2 * half];       // K = k   (lo half) / k+2 (hi half)
            a.y = A[k + 2 * half + 1];   // K = k+1 (lo half) / k+3 (hi half)
            b.x = B[k + 2 * half];
            b.y = B[k + 2 * half + 1];
            c = __builtin_amdgcn_wmma_f32_16x16x4_f32(
                    /*neg_a=*/false, a, /*neg_b=*/false, b,
                    /*c_mod=*/(short)0, c, /*reuse_a=*/false, /*reuse_b=*/false);
        }
    }
    // sigmoid + store
    #pragma unroll
    for (int i = 0; i < 8; ++i) {
        float v = c[i];
        v = 1.0f / (1.0f + __expf(-v));
        hOut[((size_t)(n0 + i + 8 * half) << 7) + (o0 + l16)] = v;
    }
}

// ---------------------------------------------------------------------------
// K3: out[b] = h[e1[b]] . (M[rel[b]] @ h[e2[b]])  — one wave per sample.
// Lane owns f in {lane, lane+32, lane+64, lane+96}; relmats are L2-resident.
// ---------------------------------------------------------------------------
__global__ __launch_bounds__(256) void k_score(const float* __restrict__ h,
                                               const float* __restrict__ M, // [R][F][F]
                                               const int* __restrict__ e1,
                                               const int* __restrict__ rel,
                                               const int* __restrict__ e2,
                                               float* __restrict__ out) {
    const int b = blockIdx.x * 8 + (threadIdx.x >> 5);
    const int lane = threadIdx.x & 31;
    if (b >= BATCH) return;

    const int r = rel[b];
    const float* __restrict__ h1 = h + ((size_t)e1[b] << 7);
    const float* __restrict__ h2 = h + ((size_t)e2[b] << 7);

    float s = 0.0f;
    #pragma unroll
    for (int fi = 0; fi < 4; ++fi) {
        const int f = lane + 32 * fi;
        const float* __restrict__ Mrow = M + ((((size_t)r << 7) + f) << 7);
        float t = 0.0f;
        for (int g = 0; g < FDIM; ++g)
            t = fmaf(Mrow[g], h2[g], t);
        s = fmaf(h1[f], t, s);
    }
    #pragma unroll
    for (int off = 16; off > 0; off >>= 1)
        s += __shfl_down(s, off, 32);
    if (lane == 0) out[b] = s;
}

// ---------------------------------------------------------------------------
// Launch: layer1 agg -> layer1 WMMA gemm -> layer2 agg -> layer2 gemm -> score
// Workspace: tmp (R*N*F f32 = 8MB) + h (N*F f32 = 2MB) = 10MB.
// h buffer is reused for both layers (k_gemm only reads tmp).
// ---------------------------------------------------------------------------
extern "C" void kernel_launch(void* const* d_in, const int* in_sizes, int n_in,
                              void* d_out, int out_size, void* d_ws, size_t ws_size,
                              hipStream_t stream) {
    const float* features = (const float*)d_in[0];
    const float* adj      = (const float*)d_in[1];
    const float* W1       = (const float*)d_in[2];
    const float* W2       = (const float*)d_in[3];
    const float* relmats  = (const float*)d_in[4];
    const int*   e1_idx   = (const int*)d_in[5];
    const int*   rel_idx  = (const int*)d_in[6];
    const int*   e2_idx   = (const int*)d_in[7];
    float* out = (float*)d_out;

    float* tmp = (float*)d_ws;                           // [R][N][F]  8MB
    float* h   = tmp + (size_t)N_REL * N_ENT * FDIM;     // [N][F]     2MB

    // Layer 1
    k_agg<<<N_REL * N_ENT, 128, 0, stream>>>(adj, features, tmp);
    k_gemm<<<N_ENT / 16, 256, 0, stream>>>(tmp, W1, h);
    // Layer 2 (reads h, overwrites tmp, then overwrites h in place)
    k_agg<<<N_REL * N_ENT, 128, 0, stream>>>(adj, h, tmp);
    k_gemm<<<N_ENT / 16, 256, 0, stream>>>(tmp, W2, h);
    // DistMult gather-scores
    k_score<<<BATCH / 8, 256, 0, stream>>>(h, relmats, e1_idx, rel_idx, e2_idx, out);
}